// LIP_AAF_Loss_70609262346794
// MI455X (gfx1250) — compile-verified
//
#include <hip/hip_runtime.h>
#include <math.h>

// ---------------- constants ----------------
#define IGN 255
#define NC 20
#define HS 80              // source H=W
#define HO 320             // output H=W
#define PPLANE (320 * 320) // pixels per plane
#define NPIX (2 * 320 * 320)
#define SEG (1 << 18)      // 262144: padded segment length for sorting
#define NSEG 25            // 20 (pred0) + 3 (hb) + 2 (fb)
#define KLD_MARGIN 3.0f
#define DEC 0.001f

typedef float v2f __attribute__((ext_vector_type(2)));
typedef float v8f __attribute__((ext_vector_type(8)));

// ---------------- helpers ----------------
__device__ __forceinline__ float blockReduceSum256(float v) {
  __shared__ float sh[256];
  int t = threadIdx.x;
  sh[t] = v;
  __syncthreads();
  for (int o = 128; o > 0; o >>= 1) {
    if (t < o) sh[t] += sh[t + o];
    __syncthreads();
  }
  float r = sh[0];
  __syncthreads();
  return r;
}

// ---------------- kernels ----------------
__global__ void k_zero(float* acc, float* segloss) {
  int t = threadIdx.x;
  if (t < 64) acc[t] = 0.f;
  if (t < 32) segloss[t] = 0.f;
}

// Bilinear (align_corners=True) interpolation matrices.
// Ry[o, i] : weight of source row i for output row o  (320x80)
// RxT[i, o]: transpose of the same matrix             (80x320)
__global__ void k_build_interp(float* __restrict__ Ry, float* __restrict__ RxT) {
  int o = blockIdx.x * blockDim.x + threadIdx.x;
  if (o >= HO) return;
  float s = (float)o * 79.0f / 319.0f;
  float fl = floorf(s);
  int i0 = (int)fl;
  int i1 = min(i0 + 1, HS - 1);
  float f = s - fl;
  for (int i = 0; i < HS; ++i) {
    float v = 0.f;
    if (i == i0) v += 1.f - f;
    if (i == i1) v += f;
    Ry[o * HS + i] = v;
    RxT[i * HO + o] = v;
  }
}

// Batched GEMM C[pl] = A[pl] (Mx K, row major, lda=K) * B[pl] (K x N, ldb=N).
// One wave (32 threads) per 16x16 output tile. Uses V_WMMA_F32_16X16X4_F32.
// A-frag layout (16x4 f32): lane(l): m=l&15, VGPR j holds K = 2*(l>>4)+j.
// B-frag layout (4x16 f32): lane(l): n=l&15, VGPR j holds K = 2*(l>>4)+j.
// C/D 16x16 f32: VGPR i holds row (l>>4)*8 + i, col l&15.
__global__ void k_gemm_wmma(const float* __restrict__ A, long long sA,
                            const float* __restrict__ B, long long sB,
                            float* __restrict__ C, long long sC, int K, int N) {
  const int lane = threadIdx.x;
  const int lo = lane & 15;
  const int hi = lane >> 4;
  const int tn = blockIdx.x * 16;
  const int tm = blockIdx.y * 16;
  const int pl = blockIdx.z;
  const float* Ap = A + (long long)pl * sA + (long long)(tm + lo) * K;
  const float* Bp = B + (long long)pl * sB + tn + lo;
  float* Cp = C + (long long)pl * sC + tn + lo;
  v8f acc = {};
  for (int k = 0; k < K; k += 4) {
    v2f a, b;
    int ka = k + 2 * hi;
    a[0] = Ap[ka];
    a[1] = Ap[ka + 1];
    b[0] = Bp[(long long)ka * N];
    b[1] = Bp[(long long)(ka + 1) * N];
    acc = __builtin_amdgcn_wmma_f32_16x16x4_f32(false, a, false, b, (short)0,
                                                acc, false, false);
  }
#pragma unroll
  for (int i = 0; i < 8; ++i)
    Cp[(long long)(tm + hi * 8 + i) * N] = acc[i];
}

// In-place channel softmax over a group of Cg channels laid out [B][Cg][H][W];
// optionally accumulates cross-entropy vs labels and the valid-pixel count.
__global__ void k_softmax_ce(float* __restrict__ buf, int Cg,
                             const int* __restrict__ labels, float* ce_acc,
                             float* valid_acc) {
  int idx = blockIdx.x * 256 + threadIdx.x; // grid covers NPIX exactly
  int b = idx / PPLANE;
  int r = idx - b * PPLANE;
  float* p = buf + (long long)b * Cg * PPLANE + r;
  float mx = -3.4e38f;
  for (int c = 0; c < Cg; ++c) mx = fmaxf(mx, p[(long long)c * PPLANE]);
  float se = 0.f;
  for (int c = 0; c < Cg; ++c) se += expf(p[(long long)c * PPLANE] - mx);
  float lse = logf(se);
  float inv = 1.f / se;
  float ce = 0.f, vc = 0.f;
  if (labels) {
    int lab = labels[idx];
    if (lab != IGN) {
      ce = -(p[(long long)lab * PPLANE] - mx - lse);
      vc = 1.f;
    }
  }
  for (int c = 0; c < Cg; ++c) {
    long long o = (long long)c * PPLANE;
    p[o] = expf(p[o] - mx) * inv;
  }
  if (ce_acc) {
    float s = blockReduceSum256(ce);
    if (threadIdx.x == 0) atomicAdd(ce_acc, s);
  }
  if (valid_acc) {
    float s = blockReduceSum256(vc);
    if (threadIdx.x == 0) atomicAdd(valid_acc, s);
  }
}

// Build sortable keys: key = (bits(err) & ~1) | fg  (err >= 0 so raw bits are
// order-monotonic). Invalid pixels and pad -> key 0 (err=0, fg=0) => they sort
// to the tail and contribute nothing, exactly as in the reference.
__global__ void k_build_keys(const float* __restrict__ prob,
                             const int* __restrict__ labels,
                             unsigned* __restrict__ keys, int Cset,
                             int planeBase, int segBase) {
  int c = blockIdx.y;
  int n = blockIdx.x * 256 + threadIdx.x; // 0..SEG-1
  unsigned key = 0u;
  if (n < NPIX) {
    int lab = labels[n];
    if (lab != IGN) {
      int b = n / PPLANE;
      int r = n - b * PPLANE;
      float p = prob[((long long)(planeBase + b * Cset + c)) * PPLANE + r];
      int fg = (lab == c) ? 1 : 0;
      float err = fabsf((float)fg - p);
      key = (__float_as_uint(err) & 0xFFFFFFFEu) | (unsigned)fg;
    }
  }
  keys[((long long)(segBase + c) << 18) + n] = key;
}

// One bitonic pass; segments are 2^18-aligned so partners never cross them.
// Overall direction: descending.
__global__ void k_bitonic(unsigned* __restrict__ keys, int size, int stride) {
  int i = blockIdx.x * 256 + threadIdx.x;
  int j = i ^ stride;
  if (j > i) {
    unsigned a = keys[i];
    unsigned b = keys[j];
    int l = i & (SEG - 1);
    bool desc = ((l & size) == 0);
    bool sw = desc ? (a < b) : (a > b);
    if (sw) {
      keys[i] = b;
      keys[j] = a;
    }
  }
}

// Segmented inclusive scan of the fg bit: stage 1 (per 1024-element chunk).
__global__ void k_scan1(const unsigned* __restrict__ keys, int* __restrict__ cums,
                        int* __restrict__ bsums) {
  __shared__ int sh[256];
  long long base = (long long)blockIdx.x * 1024;
  int t = threadIdx.x;
  int v[4];
  int s = 0;
  for (int q = 0; q < 4; ++q) {
    s += (int)(keys[base + t * 4 + q] & 1u);
    v[q] = s;
  }
  sh[t] = s;
  __syncthreads();
  for (int o = 1; o < 256; o <<= 1) {
    int x = (t >= o) ? sh[t - o] : 0;
    __syncthreads();
    sh[t] += x;
    __syncthreads();
  }
  int prev = (t > 0) ? sh[t - 1] : 0;
  for (int q = 0; q < 4; ++q) cums[base + t * 4 + q] = prev + v[q];
  if (t == 255) bsums[blockIdx.x] = sh[255];
}

// Stage 2: per-segment exclusive scan of 256 chunk sums; also emits gts.
__global__ void k_scan2(const int* __restrict__ bsums, int* __restrict__ boffs,
                        int* __restrict__ gts) {
  __shared__ int sh[256];
  int seg = blockIdx.x;
  int t = threadIdx.x;
  int own = bsums[seg * 256 + t];
  sh[t] = own;
  __syncthreads();
  for (int o = 1; o < 256; o <<= 1) {
    int x = (t >= o) ? sh[t - o] : 0;
    __syncthreads();
    sh[t] += x;
    __syncthreads();
  }
  boffs[seg * 256 + t] = sh[t] - own;
  if (t == 255) gts[seg] = sh[255];
}

// Lovasz finalize: per sorted element compute err * lovasz_grad and reduce.
__global__ void k_lovasz(const unsigned* __restrict__ keys,
                         const int* __restrict__ cums,
                         const int* __restrict__ boffs,
                         const int* __restrict__ gts,
                         float* __restrict__ segloss) {
  int gid = blockIdx.x * 256 + threadIdx.x;
  int seg = gid >> 18;
  int l = gid & (SEG - 1);
  int g = gts[seg];
  float contrib = 0.f;
  if (g > 0) {
    unsigned key = keys[gid];
    float err = __uint_as_float(key & 0xFFFFFFFEu);
    if (err > 0.f) {
      int fg = (int)(key & 1u);
      int cum = cums[gid] + boffs[gid >> 10];
      float gf = (float)g;
      float jac = 1.f - (gf - cum) / (gf + (float)(l + 1 - cum));
      float gc;
      if (l == 0) {
        gc = jac;
      } else {
        int cp = cum - fg;
        float jp = 1.f - (gf - cp) / (gf + (float)(l - cp));
        gc = jac - jp;
      }
      contrib = err * gc;
    }
  }
  float s = blockReduceSum256(contrib);
  if (threadIdx.x == 0) atomicAdd(&segloss[seg], s);
}

// Softmax the AAF class weights over the last (size) axis; keep comps 0 and 1.
// wbuf: [0..39] = we[s][c], [40..79] = wne[s][c]  (s=0 -> size1, s=1 -> size2)
__global__ void k_aaf_weights(const float* __restrict__ w_e,
                              const float* __restrict__ w_ne,
                              float* __restrict__ wbuf) {
  int c = threadIdx.x;
  if (c >= NC) return;
  float a0 = w_e[c * 3 + 0], a1 = w_e[c * 3 + 1], a2 = w_e[c * 3 + 2];
  float m = fmaxf(a0, fmaxf(a1, a2));
  float e0 = expf(a0 - m), e1 = expf(a1 - m), e2 = expf(a2 - m);
  float s = e0 + e1 + e2;
  wbuf[0 * NC + c] = e0 / s;
  wbuf[1 * NC + c] = e1 / s;
  float b0 = w_ne[c * 3 + 0], b1 = w_ne[c * 3 + 1], b2 = w_ne[c * 3 + 2];
  float m2 = fmaxf(b0, fmaxf(b1, b2));
  float f0 = expf(b0 - m2), f1 = expf(b1 - m2), f2 = expf(b2 - m2);
  float s2 = f0 + f1 + f2;
  wbuf[40 + 0 * NC + c] = f0 / s2;
  wbuf[40 + 1 * NC + c] = f1 / s2;
}

// Adaptive affinity: both sizes fused. acc[s*4 + {0:e,1:ne,2:ecnt,3:necnt}].
__global__ void k_aaf(const float* __restrict__ prob, const int* __restrict__ t0,
                      const float* __restrict__ wbuf, float* __restrict__ acc) {
  int idx = blockIdx.x * 256 + threadIdx.x;
  int b = idx / PPLANE;
  int r = idx - b * PPLANE;
  int h = r / HO;
  int w = r - h * HO;
  int la = t0[idx];
  float sums[8] = {0.f, 0.f, 0.f, 0.f, 0.f, 0.f, 0.f, 0.f};
  if (la != IGN) {
    const float* pc = prob + (long long)b * NC * PPLANE + r;
    float pcv[NC];
    for (int c = 0; c < NC; ++c) pcv[c] = pc[(long long)c * PPLANE];
    for (int s = 0; s < 2; ++s) {
      int sz = s + 1;
      float we = wbuf[s * NC + la];
      float wne = wbuf[40 + s * NC + la];
      for (int dyi = 0; dyi < 3; ++dyi) {
        for (int dxi = 0; dxi < 3; ++dxi) {
          if (dyi == 1 && dxi == 1) continue;
          int hn = h + (dyi - 1) * sz;
          int wn = w + (dxi - 1) * sz;
          if (hn < 0 || hn >= HO || wn < 0 || wn >= HO) continue;
          int ln = t0[b * PPLANE + hn * HO + wn];
          if (ln == IGN) continue;
          const float* pq = prob + (long long)b * NC * PPLANE + hn * HO + wn;
          for (int c = 0; c < NC; ++c) {
            float pv = pcv[c];
            float qv = pq[(long long)c * PPLANE];
            float p_c = fminf(fmaxf(pv, 1e-4f), 1.0f);
            float np_c = fminf(fmaxf(1.f - pv, 1e-4f), 1.0f);
            float pp = fminf(fmaxf(qv, 1e-4f), 1.0f);
            float npp = fminf(fmaxf(1.f - qv, 1e-4f), 1.0f);
            float kl = pp * logf(pp / p_c) + npp * logf(npp / np_c);
            bool edge = (c == la) != (c == ln);
            if (edge) {
              sums[s * 4 + 0] += fmaxf(0.f, KLD_MARGIN - kl) * we;
              sums[s * 4 + 2] += 1.f;
            } else {
              sums[s * 4 + 1] += kl * wne;
              sums[s * 4 + 3] += 1.f;
            }
          }
        }
      }
    }
  }
  for (int k = 0; k < 8; ++k) {
    float s = blockReduceSum256(sums[k]);
    if (threadIdx.x == 0) atomicAdd(&acc[k], s);
  }
}

__global__ void k_combine(const float* __restrict__ acc,
                          const float* __restrict__ segloss,
                          const int* __restrict__ gts, float* __restrict__ out) {
  if (threadIdx.x != 0 || blockIdx.x != 0) return;
  float l0 = 0.f;
  int p0 = 0;
  for (int c = 0; c < 20; ++c)
    if (gts[c] > 0) { l0 += segloss[c]; ++p0; }
  float lov0 = l0 / (float)max(p0, 1);
  float lh = 0.f;
  int ph = 0;
  for (int c = 20; c < 23; ++c)
    if (gts[c] > 0) { lh += segloss[c]; ++ph; }
  float lovhb = lh / (float)max(ph, 1);
  float lf = 0.f;
  int pf = 0;
  for (int c = 23; c < 25; ++c)
    if (gts[c] > 0) { lf += segloss[c]; ++pf; }
  float lovfb = lf / (float)max(pf, 1);
  float valid = fmaxf(acc[10], 1.f);
  float ce0 = acc[8] / valid;
  float cedsn = acc[9] / valid;
  float e1 = acc[0] / fmaxf(acc[2], 1.f);
  float ne1 = acc[1] / fmaxf(acc[3], 1.f);
  float e2 = acc[4] / fmaxf(acc[6], 1.f);
  float ne2 = acc[5] / fmaxf(acc[7], 1.f);
  out[0] = lov0 + (e1 + e2 + ne1 + ne2) * DEC + ce0 + 0.4f * (lovhb + lovfb) +
           0.4f * cedsn;
}

// ---------------- host ----------------
extern "C" void kernel_launch(void* const* d_in, const int* in_sizes, int n_in,
                              void* d_out, int out_size, void* d_ws,
                              size_t ws_size, hipStream_t stream) {
  (void)in_sizes; (void)n_in; (void)out_size; (void)ws_size;
  const float* pred0 = (const float*)d_in[0];
  const float* pred1 = (const float*)d_in[1];
  const float* pred2 = (const float*)d_in[2];
  const float* pdsn = (const float*)d_in[3];
  const float* w_e = (const float*)d_in[4];
  const float* w_ne = (const float*)d_in[5];
  const int* t0 = (const int*)d_in[6];
  const int* t1 = (const int*)d_in[7];
  const int* t2 = (const int*)d_in[8];
  float* out = (float*)d_out;

  char* ws = (char*)d_ws;
  size_t off = 0;
  auto alloc = [&](size_t bytes) -> char* {
    char* p = ws + off;
    off += (bytes + 255) & ~(size_t)255;
    return p;
  };
  float* Ry = (float*)alloc((size_t)HO * HS * 4);         // 320x80
  float* RxT = (float*)alloc((size_t)HS * HO * 4);        // 80x320
  float* T = (float*)alloc((size_t)90 * HS * HO * 4);     // 90 planes, 80x320
  float* L = (float*)alloc((size_t)90 * PPLANE * 4);      // logits -> probs
  unsigned* keys = (unsigned*)alloc((size_t)NSEG * SEG * 4);
  int* cums = (int*)alloc((size_t)NSEG * SEG * 4);
  int* bsums = (int*)alloc((size_t)NSEG * 256 * 4);
  int* boffs = (int*)alloc((size_t)NSEG * 256 * 4);
  int* gts = (int*)alloc(32 * 4);
  float* acc = (float*)alloc(64 * 4);
  float* segloss = (float*)alloc(32 * 4);
  float* wbuf = (float*)alloc(128 * 4);

  // 0. zero accumulators (harness does not re-zero between replays)
  k_zero<<<1, 128, 0, stream>>>(acc, segloss);

  // 1. interpolation matrices + AAF class weights
  k_build_interp<<<(HO + 63) / 64, 64, 0, stream>>>(Ry, RxT);
  k_aaf_weights<<<1, 32, 0, stream>>>(w_e, w_ne, wbuf);

  // 2. resize via WMMA GEMMs.
  //    Stage 1: T[pl] = X[pl](80x80) * RxT(80x320)
  {
    dim3 g1(HO / 16, HS / 16, 1);
    g1.z = 40; // pred0: 2*20 planes
    k_gemm_wmma<<<g1, 32, 0, stream>>>(pred0, (long long)HS * HS, RxT, 0LL,
                                       T + 0 * (size_t)HS * HO,
                                       (long long)HS * HO, HS, HO);
    g1.z = 6; // pred1: 2*3 planes
    k_gemm_wmma<<<g1, 32, 0, stream>>>(pred1, (long long)HS * HS, RxT, 0LL,
                                       T + 40 * (size_t)HS * HO,
                                       (long long)HS * HO, HS, HO);
    g1.z = 4; // pred2: 2*2 planes
    k_gemm_wmma<<<g1, 32, 0, stream>>>(pred2, (long long)HS * HS, RxT, 0LL,
                                       T + 46 * (size_t)HS * HO,
                                       (long long)HS * HO, HS, HO);
    g1.z = 40; // pred_dsn
    k_gemm_wmma<<<g1, 32, 0, stream>>>(pdsn, (long long)HS * HS, RxT, 0LL,
                                       T + 50 * (size_t)HS * HO,
                                       (long long)HS * HO, HS, HO);
  }
  //    Stage 2: L[pl] = Ry(320x80) * T[pl](80x320), all 90 planes batched
  {
    dim3 g2(HO / 16, HO / 16, 90);
    k_gemm_wmma<<<g2, 32, 0, stream>>>(Ry, 0LL, T, (long long)HS * HO, L,
                                       (long long)PPLANE, HS, HO);
  }

  // 3. softmax (in place) + cross-entropy accumulation
  const int PXB = NPIX / 256; // 800 blocks
  k_softmax_ce<<<PXB, 256, 0, stream>>>(L + 0 * (size_t)PPLANE, 20, t0,
                                        &acc[8], &acc[10]); // pred0: CE + valid
  k_softmax_ce<<<PXB, 256, 0, stream>>>(L + 40 * (size_t)PPLANE, 3, nullptr,
                                        nullptr, nullptr);  // hb
  k_softmax_ce<<<PXB, 256, 0, stream>>>(L + 46 * (size_t)PPLANE, 2, nullptr,
                                        nullptr, nullptr);  // fb
  k_softmax_ce<<<PXB, 256, 0, stream>>>(L + 50 * (size_t)PPLANE, 20, t0,
                                        &acc[9], nullptr);  // dsn: CE only

  // 4. Lovasz: build keys for the 3 probability sets
  {
    dim3 gk(SEG / 256, 20);
    k_build_keys<<<gk, 256, 0, stream>>>(L, t0, keys, 20, 0, 0);
    gk.y = 3;
    k_build_keys<<<gk, 256, 0, stream>>>(L, t1, keys, 3, 40, 20);
    gk.y = 2;
    k_build_keys<<<gk, 256, 0, stream>>>(L, t2, keys, 2, 46, 23);
  }

  // 5. segmented descending bitonic sort (171 passes)
  {
    const int nblk = NSEG * SEG / 256;
    for (int size = 2; size <= SEG; size <<= 1)
      for (int stride = size >> 1; stride > 0; stride >>= 1)
        k_bitonic<<<nblk, 256, 0, stream>>>(keys, size, stride);
  }

  // 6. segmented scan of fg + Lovasz gradient reduction
  k_scan1<<<NSEG * SEG / 1024, 256, 0, stream>>>(keys, cums, bsums);
  k_scan2<<<NSEG, 256, 0, stream>>>(bsums, boffs, gts);
  k_lovasz<<<NSEG * SEG / 256, 256, 0, stream>>>(keys, cums, boffs, gts,
                                                 segloss);

  // 7. adaptive affinity (sizes 1 and 2 fused)
  k_aaf<<<PXB, 256, 0, stream>>>(L, t0, wbuf, acc);

  // 8. final scalar combine
  k_combine<<<1, 32, 0, stream>>>(acc, segloss, gts, out);
}